// OneHopGCNNormNodeLabelAggregator_2568390443269
// MI455X (gfx1250) — compile-verified
//
#include <hip/hip_runtime.h>
#include <hip/hip_bf16.h>

// CDNA5 WMMA vector types (wave32): 16x4 f32 A = 2 VGPRs, 4x16 f32 B = 2 VGPRs,
// 16x16 f32 C/D = 8 VGPRs.
typedef __attribute__((ext_vector_type(2))) float v2f;
typedef __attribute__((ext_vector_type(8))) float v8f;

// ---------------------------------------------------------------------------
// Pass 0: zero integer scratch (deg + tileCount + tileCursor are contiguous).
// ---------------------------------------------------------------------------
__global__ void zero_ints_kernel(int* __restrict__ p, int n) {
    int i = blockIdx.x * blockDim.x + threadIdx.x;
    if (i < n) p[i] = 0;
}

// ---------------------------------------------------------------------------
// Pass 1: deg[src[e]] += 1 over all edges (row-sum degrees, matches reference).
// ---------------------------------------------------------------------------
__global__ void deg_count_kernel(const long long* __restrict__ ei,
                                 int* __restrict__ deg, int E) {
    int e = blockIdx.x * blockDim.x + threadIdx.x;
    if (e < E) atomicAdd(&deg[(int)ei[e]], 1);
}

// ---------------------------------------------------------------------------
// Pass 2: dis[i] = deg>0 ? rsqrt(deg) : 0.
// ---------------------------------------------------------------------------
__global__ void deg_inv_sqrt_kernel(const int* __restrict__ deg,
                                    float* __restrict__ dis, int N) {
    int i = blockIdx.x * blockDim.x + threadIdx.x;
    if (i < N) {
        int d = deg[i];
        dis[i] = (d > 0) ? rsqrtf((float)d) : 0.0f;
    }
}

// ---------------------------------------------------------------------------
// Pass 3: count edges per 16-row destination tile (only dst < batch matter).
// ---------------------------------------------------------------------------
__global__ void tile_count_kernel(const long long* __restrict__ ei,
                                  int* __restrict__ tcnt, int E, int B) {
    int e = blockIdx.x * blockDim.x + threadIdx.x;
    if (e < E) {
        int d = (int)ei[(size_t)E + e];
        if (d < B) atomicAdd(&tcnt[d >> 4], 1);
    }
}

// ---------------------------------------------------------------------------
// Pass 4: exclusive scan over tile counts (nT threads, nT power of two).
// ---------------------------------------------------------------------------
__global__ void scan_tiles_kernel(const int* __restrict__ tcnt,
                                  int* __restrict__ off, int n) {
    __shared__ int tmp[512];
    int tid = threadIdx.x;
    int v = (tid < n) ? tcnt[tid] : 0;
    tmp[tid] = v;
    __syncthreads();
    for (int d = 1; d < n; d <<= 1) {
        int t = tmp[tid];
        int u = (tid >= d) ? tmp[tid - d] : 0;
        __syncthreads();
        tmp[tid] = t + u;
        __syncthreads();
    }
    if (tid < n) off[tid] = tmp[tid] - v;       // exclusive
    if (tid == n - 1) off[n] = tmp[tid];        // total
}

// ---------------------------------------------------------------------------
// Pass 5: scatter edge ids into per-tile buckets.
// ---------------------------------------------------------------------------
__global__ void bucket_fill_kernel(const long long* __restrict__ ei,
                                   const int* __restrict__ off,
                                   int* __restrict__ cursor,
                                   int* __restrict__ buckets, int E, int B) {
    int e = blockIdx.x * blockDim.x + threadIdx.x;
    if (e < E) {
        int d = (int)ei[(size_t)E + e];
        if (d < B) {
            int t = d >> 4;
            int p = atomicAdd(&cursor[t], 1);
            buckets[off[t] + p] = e;
        }
    }
}

// ---------------------------------------------------------------------------
// Pass 6: WMMA aggregation. One block per 16-row dst tile; 8 waves x 2 column
// tiles cover all 256 feature columns. For each chunk of 4 bucketed edges:
//   A[16x4]  : A[m][k] = (dst[e_k]==base+m) ? w[e_k] : 0   (selector weights,
//              built BRANCHLESSLY via sign-mask AND so the ds loads stay
//              unconditional and mergeable -> no exec-mask juggling in loop)
//   B[4x16]  : gathered x[src[e_k]] feature slab
//   C       += A x B  via V_WMMA_F32_16X16X4_F32 (rank-4 update)
// Unrolled x2 (8 edges / iter): 4 WMMAs + 8 clause-able gather loads per
// iteration to hide WMMA latency. No float atomics anywhere; each tile's 16
// output rows are owned exclusively.
// ---------------------------------------------------------------------------
__global__ void agg_wmma_kernel(const float* __restrict__ x,
                                const long long* __restrict__ ei,
                                const float* __restrict__ dis,
                                const int* __restrict__ off,
                                const int* __restrict__ buckets,
                                float* __restrict__ out, int E) {
    __shared__ int   srcS[256];
    __shared__ int   dstS[256];
    __shared__ float wS[256];

    const int t    = blockIdx.x;
    const int base = t << 4;            // first dst row of this tile
    const int tid  = threadIdx.x;
    const int lane = tid & 31;
    const int wave = tid >> 5;          // 0..7

    const int start = off[t];
    const int nE    = off[t + 1] - start;

    const int m   = lane & 15;          // A row / B col / C col
    const int kb  = (lane >> 4) << 1;   // K base: 0 or 2
    const int cb0 = wave << 4;          // feature column tiles: wave, wave+8
    const int cb1 = (wave + 8) << 4;
    const int tgt = base + m;           // dst row this lane selects for

    // Per-lane column base pointers: per-load address math reduces to
    // (src * 1024B) shift-add onto these.
    const float* __restrict__ xc0 = x + cb0 + m;
    const float* __restrict__ xc1 = x + cb1 + m;

    v8f acc0 = {};
    v8f acc1 = {};

    for (int sb = 0; sb < nE; sb += 256) {
        // Stage up to 256 edges into LDS; invalid slots get w=0 / dst=-1 so
        // they contribute exactly zero through the selector matrix.
        int e = sb + tid;
        if (e < nE) {
            int eid = buckets[start + e];
            int s   = (int)ei[eid];
            int d   = (int)ei[(size_t)E + eid];
            srcS[tid] = s;
            dstS[tid] = d;
            wS[tid]   = dis[s] * dis[d];
            __builtin_prefetch(&x[(size_t)s * 256], 0, 3);  // global_prefetch_b8
        } else {
            srcS[tid] = 0;
            dstS[tid] = -1;
            wS[tid]   = 0.0f;
        }
        __syncthreads();

        const int cnt = (nE - sb < 256) ? (nE - sb) : 256;
        for (int c8 = 0; c8 < cnt; c8 += 8) {   // uniform: EXEC all-1 at WMMA
#pragma unroll
            for (int h = 0; h < 2; ++h) {
                const int e0 = c8 + (h << 2) + kb;
                const int e1 = e0 + 1;

                // Unconditional LDS reads (e0/e1 adjacent -> b64 mergeable).
                const int   d0 = dstS[e0];
                const int   d1 = dstS[e1];
                const float w0 = wS[e0];
                const float w1 = wS[e1];
                const int   s0 = srcS[e0];
                const int   s1 = srcS[e1];

                // Branchless selector: w & (-(d==tgt)).
                const int msk0 = -(int)(d0 == tgt);
                const int msk1 = -(int)(d1 == tgt);
                v2f a;
                a.x = __int_as_float(__float_as_int(w0) & msk0);
                a.y = __int_as_float(__float_as_int(w1) & msk1);

                const size_t r0 = (size_t)s0 << 8;   // row offsets in floats
                const size_t r1 = (size_t)s1 << 8;
                v2f b0, b1;
                b0.x = xc0[r0];
                b0.y = xc0[r1];
                b1.x = xc1[r0];
                b1.y = xc1[r1];

                acc0 = __builtin_amdgcn_wmma_f32_16x16x4_f32(
                    false, a, false, b0, (short)0, acc0, false, false);
                acc1 = __builtin_amdgcn_wmma_f32_16x16x4_f32(
                    false, a, false, b1, (short)0, acc1, false, false);
            }
        }
        __syncthreads();
    }

    // Store C tiles into the second half (cols 256..511) of the output rows.
    // C layout: VGPR j -> M = j + 8*(lane>=16), N = lane&15.
    const int rofs = (lane >> 4) << 3;  // 0 or 8
    for (int j = 0; j < 8; ++j) {
        int row = base + rofs + j;
        out[(size_t)row * 512 + 256 + cb0 + m] = acc0[j];
        out[(size_t)row * 512 + 256 + cb1 + m] = acc1[j];
    }
}

// ---------------------------------------------------------------------------
// Pass 7: out[:B, :256] = x[:B, :]  (float4-vectorized copy).
// ---------------------------------------------------------------------------
__global__ void copy_x_kernel(const float* __restrict__ x,
                              float* __restrict__ out, int B) {
    int i = blockIdx.x * blockDim.x + threadIdx.x;   // over B*64 float4s
    int total = B * 64;
    if (i < total) {
        int row = i >> 6;
        int c   = i & 63;
        const float4* xv = (const float4*)x;
        float4*       ov = (float4*)out;
        ov[(size_t)row * 128 + c] = xv[(size_t)row * 64 + c];
    }
}

// ---------------------------------------------------------------------------
extern "C" void kernel_launch(void* const* d_in, const int* in_sizes, int n_in,
                              void* d_out, int out_size, void* d_ws, size_t ws_size,
                              hipStream_t stream) {
    const float*     x  = (const float*)d_in[0];
    const long long* ei = (const long long*)d_in[1];  // int64 [2, E] row-major

    const int D  = 256;
    const int N  = in_sizes[0] / D;        // 20000
    const int E  = in_sizes[1] / 2;        // 320000
    const int B  = out_size / (2 * D);     // 4096
    const int nT = B / 16;                 // 256 dst tiles

    // Workspace layout (all 4-byte typed, contiguous int block first so one
    // zeroing launch covers deg + tileCount + tileCursor).
    int*   deg     = (int*)d_ws;           // [N]
    int*   tcnt    = deg + N;              // [nT]
    int*   cursor  = tcnt + nT;            // [nT]
    int*   off     = cursor + nT;          // [nT+1]
    float* dis     = (float*)(off + nT + 1); // [N]
    int*   buckets = (int*)(dis + N);      // [E] worst case

    float* out = (float*)d_out;

    const int TB = 256;
    const int zn = N + 2 * nT;
    zero_ints_kernel<<<(zn + TB - 1) / TB, TB, 0, stream>>>(deg, zn);
    deg_count_kernel<<<(E + TB - 1) / TB, TB, 0, stream>>>(ei, deg, E);
    deg_inv_sqrt_kernel<<<(N + TB - 1) / TB, TB, 0, stream>>>(deg, dis, N);
    tile_count_kernel<<<(E + TB - 1) / TB, TB, 0, stream>>>(ei, tcnt, E, B);
    scan_tiles_kernel<<<1, nT, 0, stream>>>(tcnt, off, nT);
    bucket_fill_kernel<<<(E + TB - 1) / TB, TB, 0, stream>>>(ei, off, cursor,
                                                             buckets, E, B);
    agg_wmma_kernel<<<nT, TB, 0, stream>>>(x, ei, dis, off, buckets, out, E);
    copy_x_kernel<<<(B * 64 + TB - 1) / TB, TB, 0, stream>>>(x, out, B);
}